// KnnLossSeq_50775103373766
// MI455X (gfx1250) — compile-verified
//
#include <hip/hip_runtime.h>
#include <math.h>

#define NPTS   16384
#define KNN    16
#define NSEQ   8
#define RADIUS2 1.0f

typedef __attribute__((ext_vector_type(2))) float v2f;
typedef __attribute__((ext_vector_type(4))) float v4f;
typedef __attribute__((ext_vector_type(8))) float v8f;

// One wave (32 lanes) per block; block owns 32 query rows.
// Gram matrix tiles computed with V_WMMA_F32_16X16X4_F32 (K padded 3->4).
__global__ __launch_bounds__(32) void knn_loss_main(
    const float* __restrict__ pc,      // (NPTS, 3)
    const float* __restrict__ flow,    // (NSEQ, NPTS, 3)
    const float* __restrict__ w,       // (NPTS)
    float* __restrict__ partial)       // (gridDim.x)
{
  __shared__ float lds[32 * 16];

  const int lane = threadIdx.x;       // 0..31
  const int i0   = blockIdx.x * 32;

  // ---- A-matrix tiles (16x4 f32 layout):
  // lane l: M = l&15, Kbase = (l>>4)*2 ; VGPR0 = K=Kbase, VGPR1 = K=Kbase+1
  const int  am = lane & 15;
  const bool lo = (lane < 16);        // Kbase==0 half vs Kbase==2 half

  v2f a0, a1;
  {
    const float* r0 = pc + (size_t)(i0 + am) * 3;
    const float* r1 = pc + (size_t)(i0 + 16 + am) * 3;
    const float x0 = r0[0], y0 = r0[1], z0 = r0[2];   // one b96 each
    const float x1 = r1[0], y1 = r1[1], z1 = r1[2];
    a0.x = lo ? x0 : z0;  a0.y = lo ? y0 : 0.0f;      // coord 3 = zero pad
    a1.x = lo ? x1 : z1;  a1.y = lo ? y1 : 0.0f;
  }

  // |p_i|^2 for this lane's own row
  float x2i;
  {
    const float* r = pc + (size_t)(i0 + lane) * 3;
    x2i = r[0]*r[0] + r[1]*r[1] + r[2]*r[2];
  }

  // top-KNN sorted ascending (d2, idx)
  float bd[KNN];
  int   bi[KNN];
#pragma unroll
  for (int k = 0; k < KNN; ++k) { bd[k] = 3.0e38f; bi[k] = 0; }

  for (int j0 = 0; j0 < NPTS; j0 += 16) {
    if (j0 + 16 < NPTS)
      __builtin_prefetch(pc + (size_t)(j0 + 16) * 3, 0, 1);

    // ---- B-matrix tile (4x16 f32): lane l: N = l&15, Kbase = (l>>4)*2.
    // Single b96 load per lane; build fragment with selects (no divergence).
    const int jn = j0 + am;
    const float* q = pc + (size_t)jn * 3;
    const float qx = q[0], qy = q[1], qz = q[2];
    v2f b;
    b.x = lo ? qx : qz;
    b.y = lo ? qy : 0.0f;
    const float x2j = qx*qx + qy*qy + qz*qz;

    v8f cz = {};
    // D = A x B (+0): dot products of point pairs
    v8f c0 = __builtin_amdgcn_wmma_f32_16x16x4_f32(
        false, a0, false, b, (short)0, cz, false, false);
    v8f c1 = __builtin_amdgcn_wmma_f32_16x16x4_f32(
        false, a1, false, b, (short)0, cz, false, false);

    // C/D layout: lane holds N = lane&15; VGPR v holds M = v + 8*(lane>>4).
    // Store partial d2 (x2j - 2*dot); row-owner lane adds its x2i.
    const int mbase = lo ? 0 : 8;
#pragma unroll
    for (int v = 0; v < 8; ++v) {
      lds[(mbase + v)      * 16 + am] = fmaf(-2.0f, c0[v], x2j);
      lds[(mbase + v + 16) * 16 + am] = fmaf(-2.0f, c1[v], x2j);
    }
    __syncthreads();

    // ---- selection: lane owns row (i0+lane). Pull all 16 candidates with
    // four ds_load_b128 (64B-aligned row), then pure-VALU insertion.
    float dv[16];
    {
      const v4f* row = (const v4f*)&lds[lane * 16];
      v4f r0 = row[0], r1 = row[1], r2 = row[2], r3 = row[3];
#pragma unroll
      for (int t = 0; t < 4; ++t) {
        dv[0 + t]  = r0[t];
        dv[4 + t]  = r1[t];
        dv[8 + t]  = r2[t];
        dv[12 + t] = r3[t];
      }
    }

#pragma unroll
    for (int n = 0; n < 16; ++n) {
      const float d2 = fmaxf(x2i + dv[n], 0.0f);
      const int   j  = j0 + n;
      if (d2 < bd[KNN - 1]) {
        // branchless sorted insert (static indices only; ties keep lower j)
#pragma unroll
        for (int k = KNN - 1; k >= 0; --k) {
          const float up  = (k > 0) ? bd[k - 1] : -1.0f; // sentinel <= d2
          const int   upi = (k > 0) ? bi[k - 1] : j;
          if (bd[k] > d2) {
            const bool shift = (up > d2);
            bd[k] = shift ? up  : d2;
            bi[k] = shift ? upi : j;
          }
        }
      }
    }
    __syncthreads();
  }

  // ---- radius clamp: dist > 1 (d2 > 1) -> use nearest index (self)
  int nidx[KNN];
#pragma unroll
  for (int k = 0; k < KNN; ++k)
    nidx[k] = (bd[k] > RADIUS2) ? bi[0] : bi[k];

  // ---- fused loss for this lane's point
  const int i = i0 + lane;
  float fx[NSEQ], fy[NSEQ], fz[NSEQ];
#pragma unroll
  for (int s = 0; s < NSEQ; ++s) {
    const float* f = flow + ((size_t)s * NPTS + i) * 3;
    fx[s] = f[0]; fy[s] = f[1]; fz[s] = f[2];
  }

  float acc = 0.0f;
  for (int k = 0; k < KNN; ++k) {
    const int j = nidx[k];
#pragma unroll
    for (int s = 0; s < NSEQ; ++s) {
      const float* g = flow + ((size_t)s * NPTS + j) * 3;
      const float dx = fx[s] - g[0];
      const float dy = fy[s] - g[1];
      const float dz = fz[s] - g[2];
      const float sq = dx*dx + dy*dy + dz*dz;
      acc += sqrtf(sq);                  // sqrt(0)=0 matches where(sq>0,...)
    }
  }
  const float contrib = w[i] * acc * (1.0f / KNN);

  // ---- deterministic single-wave reduction -> per-block partial
  lds[lane] = contrib;
  __syncthreads();
  if (lane == 0) {
    float s = 0.0f;
#pragma unroll
    for (int t = 0; t < 32; ++t) s += lds[t];
    partial[blockIdx.x] = s;
  }
}

// Deterministic finalize: sum 512 block partials + weight sum, emit scalar.
__global__ __launch_bounds__(256) void knn_loss_finalize(
    const float* __restrict__ partial, int nPartial,
    const float* __restrict__ w, int n,
    float* __restrict__ out)
{
  __shared__ float sl[256];
  __shared__ float sw[256];
  const int tid = threadIdx.x;
  float a = 0.0f, b = 0.0f;
  for (int i = tid; i < nPartial; i += 256) a += partial[i];
  for (int i = tid; i < n; i += 256)        b += w[i];
  sl[tid] = a; sw[tid] = b;
  __syncthreads();
  for (int off = 128; off > 0; off >>= 1) {
    if (tid < off) { sl[tid] += sl[tid + off]; sw[tid] += sw[tid + off]; }
    __syncthreads();
  }
  if (tid == 0) {
    const float loss = sl[0] * (1.0f / NSEQ);
    const float wsum = sw[0];
    out[0] = (wsum > 0.0f) ? (loss / wsum) : loss;
  }
}

extern "C" void kernel_launch(void* const* d_in, const int* in_sizes, int n_in,
                              void* d_out, int out_size, void* d_ws, size_t ws_size,
                              hipStream_t stream) {
  const float* pc   = (const float*)d_in[0];   // (1, 16384, 3) f32
  const float* flow = (const float*)d_in[1];   // (8, 16384, 3) f32
  const float* w    = (const float*)d_in[2];   // (16384,)      f32
  float* out     = (float*)d_out;              // scalar f32
  float* partial = (float*)d_ws;               // 512 floats of scratch

  const int nBlocks = NPTS / 32;               // 512
  knn_loss_main<<<nBlocks, 32, 0, stream>>>(pc, flow, w, partial);
  knn_loss_finalize<<<1, 256, 0, stream>>>(partial, nBlocks, w, NPTS, out);
}